// GATModel_74972949119166
// MI455X (gfx1250) — compile-verified
//
#include <hip/hip_runtime.h>

#define N_NODES 50000
#define N_EDGES 800000
#define ET (N_EDGES + N_NODES)   // with self-loops
#define IN_CH 128
#define HID_CH 64
#define OUT_CH 2
#define HEADS1 4
#define OUTW1 (HEADS1 * HID_CH)  // 256
#define SLOPE 0.2f

typedef float v2f __attribute__((ext_vector_type(2)));
typedef float v8f __attribute__((ext_vector_type(8)));

__device__ __forceinline__ float leaky(float v) { return v >= 0.f ? v : SLOPE * v; }

// monotonic float <-> uint mapping so atomicMax(u32) implements float max;
// 0x00000000 is below ford(-inf) so memset(0) is a safe init.
__device__ __forceinline__ unsigned ford(float f) {
  unsigned u = __float_as_uint(f);
  return (u & 0x80000000u) ? ~u : (u | 0x80000000u);
}
__device__ __forceinline__ float forddec(unsigned u) {
  return (u & 0x80000000u) ? __uint_as_float(u & 0x7fffffffu) : __uint_as_float(~u);
}

// ---------------------------------------------------------------------------
// Layer-1 GEMM: H1[N,256] = X[N,128] @ W1[128,256], exact f32 via WMMA.
// One wave computes a 16x64 output block (4 accumulators): the A fragment is
// loaded once per K-step (one b64 per lane) and reused by 4 WMMAs, cutting
// redundant X traffic 4x vs one-tile-per-wave.
// ---------------------------------------------------------------------------
__global__ void gemm1_wmma_kernel(const float* __restrict__ X,
                                  const float* __restrict__ W,
                                  float* __restrict__ Hout) {
  const int NB_N = OUTW1 / 64;                 // 4 column blocks of 64
  const int wid  = (blockIdx.x * blockDim.x + threadIdx.x) >> 5;
  const int lane = threadIdx.x & 31;
  const int tile_m = wid / NB_N;
  const int nb     = wid - tile_m * NB_N;
  if (tile_m >= N_NODES / 16) return;          // wave-uniform exit
  const int m0 = tile_m * 16;
  const int n0 = nb * 64;
  const int hi = lane >> 4;                    // lane half: 0 -> K{0,1}, 1 -> K{2,3}
  const int l  = lane & 15;

  v8f acc[4];
#pragma unroll
  for (int j = 0; j < 4; ++j) acc[j] = (v8f){0.f, 0.f, 0.f, 0.f, 0.f, 0.f, 0.f, 0.f};

  const float* xrow = X + (size_t)(m0 + l) * IN_CH;
#pragma unroll
  for (int k = 0; k < IN_CH; k += 4) {
    const int ka = k + 2 * hi;
    v2f a;
    a.x = xrow[ka];                            // A: 16x4, VGPR0/1 = K, K+1 per half
    a.y = xrow[ka + 1];
#pragma unroll
    for (int j = 0; j < 4; ++j) {              // reuse A across 4 N-tiles
      v2f b;
      b.x = W[(size_t)ka * OUTW1 + n0 + 16 * j + l];
      b.y = W[(size_t)(ka + 1) * OUTW1 + n0 + 16 * j + l];
      acc[j] = __builtin_amdgcn_wmma_f32_16x16x4_f32(false, a, false, b,
                                                     (short)0, acc[j], false, false);
    }
  }
#pragma unroll
  for (int j = 0; j < 4; ++j)
#pragma unroll
    for (int r = 0; r < 8; ++r)                // D: VGPR r -> row (hi*8 + r)
      Hout[(size_t)(m0 + hi * 8 + r) * OUTW1 + n0 + 16 * j + l] = acc[j][r];
}

// ---------------------------------------------------------------------------
// attention logits per node/head: as1[n,h] = h1[n,h,:].a_src[h,:], same for dst
// ---------------------------------------------------------------------------
__global__ void attdot1_kernel(const float* __restrict__ H1,
                               const float* __restrict__ a_src,
                               const float* __restrict__ a_dst,
                               float* __restrict__ as1, float* __restrict__ ad1) {
  int t = blockIdx.x * blockDim.x + threadIdx.x;
  if (t >= N_NODES * HEADS1) return;
  int n = t >> 2, h = t & 3;
  const float* hp = H1 + (size_t)n * OUTW1 + h * HID_CH;
  const float* sp = a_src + h * HID_CH;
  const float* dp = a_dst + h * HID_CH;
  float ds = 0.f, dd = 0.f;
#pragma unroll 8
  for (int c = 0; c < HID_CH; ++c) { float v = hp[c]; ds += v * sp[c]; dd += v * dp[c]; }
  as1[t] = ds;
  ad1[t] = dd;
}

// ---------------------------------------------------------------------------
// segment max / segment sum over incoming edges (templated on #heads)
// ---------------------------------------------------------------------------
template <int H>
__global__ void edge_max_kernel(const int* __restrict__ ei,
                                const float* __restrict__ as_,
                                const float* __restrict__ ad_,
                                unsigned* __restrict__ m) {
  int t = blockIdx.x * blockDim.x + threadIdx.x;
  if (t >= ET) return;
  int s, d;
  if (t < N_EDGES) { s = ei[t]; d = ei[N_EDGES + t]; } else { s = d = t - N_EDGES; }
#pragma unroll
  for (int h = 0; h < H; ++h) {
    float e = leaky(as_[s * H + h] + ad_[d * H + h]);
    atomicMax(&m[d * H + h], ford(e));
  }
}

template <int H>
__global__ void edge_sum_kernel(const int* __restrict__ ei,
                                const float* __restrict__ as_,
                                const float* __restrict__ ad_,
                                const unsigned* __restrict__ m,
                                float* __restrict__ z) {
  int t = blockIdx.x * blockDim.x + threadIdx.x;
  if (t >= ET) return;
  int s, d;
  if (t < N_EDGES) { s = ei[t]; d = ei[N_EDGES + t]; } else { s = d = t - N_EDGES; }
#pragma unroll
  for (int h = 0; h < H; ++h) {
    float e = leaky(as_[s * H + h] + ad_[d * H + h]);
    atomicAdd(&z[d * H + h], __expf(e - forddec(m[d * H + h])));
  }
}

// ---------------------------------------------------------------------------
// layer-1 aggregation: one wave per edge; head-mean (x0.25) folded into the
// scatter-add so only 64 atomics/edge instead of 256. Edge id forced uniform
// so index fetches and alpha math scalarize to SGPR/SALU.
// ---------------------------------------------------------------------------
__global__ void edge_agg1_kernel(const int* __restrict__ ei,
                                 const float* __restrict__ as_,
                                 const float* __restrict__ ad_,
                                 const unsigned* __restrict__ m,
                                 const float* __restrict__ z,
                                 const float* __restrict__ H1,
                                 float* __restrict__ acc1) {
  int wid  = __builtin_amdgcn_readfirstlane((blockIdx.x * blockDim.x + threadIdx.x) >> 5);
  int lane = threadIdx.x & 31;
  if (wid >= ET) return;
  int s, d;
  if (wid < N_EDGES) {
    s = __builtin_amdgcn_readfirstlane(ei[wid]);
    d = __builtin_amdgcn_readfirstlane(ei[N_EDGES + wid]);
  } else {
    s = d = wid - N_EDGES;
  }
  float alpha[HEADS1];
#pragma unroll
  for (int h = 0; h < HEADS1; ++h) {
    float e = leaky(as_[s * HEADS1 + h] + ad_[d * HEADS1 + h]);
    float p = __expf(e - forddec(m[d * HEADS1 + h]));
    alpha[h] = p / z[d * HEADS1 + h];
  }
#pragma unroll
  for (int part = 0; part < 2; ++part) {
    int c = lane + part * 32;
    float a = 0.f;
#pragma unroll
    for (int h = 0; h < HEADS1; ++h)
      a += alpha[h] * H1[(size_t)s * OUTW1 + h * HID_CH + c];
    atomicAdd(&acc1[(size_t)d * HID_CH + c], 0.25f * a);
  }
}

// in-place: act = relu(acc + b1)
__global__ void relu_bias1_kernel(float* __restrict__ acc1, const float* __restrict__ b1) {
  int t = blockIdx.x * blockDim.x + threadIdx.x;
  if (t >= N_NODES * HID_CH) return;
  float v = acc1[t] + b1[t & (HID_CH - 1)];
  acc1[t] = v > 0.f ? v : 0.f;
}

// ---------------------------------------------------------------------------
// layer-2 linear + attention logits (N=2 columns: VALU dot-64 per node)
// ---------------------------------------------------------------------------
__global__ void layer2_lin_kernel(const float* __restrict__ act1,
                                  const float* __restrict__ W2,
                                  const float* __restrict__ a_src2,
                                  const float* __restrict__ a_dst2,
                                  float* __restrict__ h2,
                                  float* __restrict__ as2,
                                  float* __restrict__ ad2) {
  int n = blockIdx.x * blockDim.x + threadIdx.x;
  if (n >= N_NODES) return;
  const float* ap = act1 + (size_t)n * HID_CH;
  float o0 = 0.f, o1 = 0.f;
#pragma unroll 8
  for (int c = 0; c < HID_CH; ++c) {
    float v = ap[c];
    o0 += v * W2[c * 2 + 0];
    o1 += v * W2[c * 2 + 1];
  }
  h2[n * 2 + 0] = o0;
  h2[n * 2 + 1] = o1;
  as2[n] = o0 * a_src2[0] + o1 * a_src2[1];
  ad2[n] = o0 * a_dst2[0] + o1 * a_dst2[1];
}

__global__ void edge_agg2_kernel(const int* __restrict__ ei,
                                 const float* __restrict__ as2,
                                 const float* __restrict__ ad2,
                                 const unsigned* __restrict__ m2,
                                 const float* __restrict__ z2,
                                 const float* __restrict__ h2,
                                 float* __restrict__ out) {
  int t = blockIdx.x * blockDim.x + threadIdx.x;
  if (t >= ET) return;
  int s, d;
  if (t < N_EDGES) { s = ei[t]; d = ei[N_EDGES + t]; } else { s = d = t - N_EDGES; }
  float e = leaky(as2[s] + ad2[d]);
  float alpha = __expf(e - forddec(m2[d])) / z2[d];
  atomicAdd(&out[d * 2 + 0], alpha * h2[s * 2 + 0]);
  atomicAdd(&out[d * 2 + 1], alpha * h2[s * 2 + 1]);
}

__global__ void bias2_kernel(float* __restrict__ out, const float* __restrict__ b2) {
  int t = blockIdx.x * blockDim.x + threadIdx.x;
  if (t >= N_NODES * OUT_CH) return;
  out[t] += b2[t & 1];
}

// ---------------------------------------------------------------------------
extern "C" void kernel_launch(void* const* d_in, const int* in_sizes, int n_in,
                              void* d_out, int out_size, void* d_ws, size_t ws_size,
                              hipStream_t stream) {
  (void)in_sizes; (void)n_in; (void)out_size; (void)ws_size;
  const float* x      = (const float*)d_in[0];
  const int*   ei     = (const int*)d_in[1];
  const float* W1     = (const float*)d_in[2];
  const float* a_src1 = (const float*)d_in[3];
  const float* a_dst1 = (const float*)d_in[4];
  const float* b1     = (const float*)d_in[5];
  const float* W2     = (const float*)d_in[6];
  const float* a_src2 = (const float*)d_in[7];
  const float* a_dst2 = (const float*)d_in[8];
  const float* b2     = (const float*)d_in[9];
  float* out = (float*)d_out;

  // workspace carve-out (256B aligned)
  char* ws = (char*)d_ws;
  size_t off = 0;
  auto carve = [&](size_t bytes) -> char* {
    char* p = ws + off;
    off += (bytes + 255) & ~(size_t)255;
    return p;
  };
  float*    h1   = (float*)carve(sizeof(float) * (size_t)N_NODES * OUTW1);   // 51.2 MB
  float*    as1  = (float*)carve(sizeof(float) * (size_t)N_NODES * HEADS1);
  float*    ad1  = (float*)carve(sizeof(float) * (size_t)N_NODES * HEADS1);
  unsigned* m1   = (unsigned*)carve(sizeof(unsigned) * (size_t)N_NODES * HEADS1);
  float*    z1   = (float*)carve(sizeof(float) * (size_t)N_NODES * HEADS1);
  float*    acc1 = (float*)carve(sizeof(float) * (size_t)N_NODES * HID_CH);  // 12.8 MB
  float*    h2   = (float*)carve(sizeof(float) * (size_t)N_NODES * OUT_CH);
  float*    as2  = (float*)carve(sizeof(float) * (size_t)N_NODES);
  float*    ad2  = (float*)carve(sizeof(float) * (size_t)N_NODES);
  unsigned* m2   = (unsigned*)carve(sizeof(unsigned) * (size_t)N_NODES);
  float*    z2   = (float*)carve(sizeof(float) * (size_t)N_NODES);

  const int B = 256;  // 8 wave32 per block

  // deterministic zero-init (graph-capture-safe)
  hipMemsetAsync(m1,   0, sizeof(unsigned) * (size_t)N_NODES * HEADS1, stream);
  hipMemsetAsync(z1,   0, sizeof(float)    * (size_t)N_NODES * HEADS1, stream);
  hipMemsetAsync(acc1, 0, sizeof(float)    * (size_t)N_NODES * HID_CH, stream);
  hipMemsetAsync(m2,   0, sizeof(unsigned) * (size_t)N_NODES, stream);
  hipMemsetAsync(z2,   0, sizeof(float)    * (size_t)N_NODES, stream);
  hipMemsetAsync(out,  0, sizeof(float)    * (size_t)N_NODES * OUT_CH, stream);

  // ---- layer 1 ----
  {
    int waves = (N_NODES / 16) * (OUTW1 / 64);          // 12500 waves (16x64 blocks)
    int grid  = (waves * 32 + B - 1) / B;
    gemm1_wmma_kernel<<<grid, B, 0, stream>>>(x, W1, h1);
  }
  attdot1_kernel<<<(N_NODES * HEADS1 + B - 1) / B, B, 0, stream>>>(h1, a_src1, a_dst1, as1, ad1);
  edge_max_kernel<HEADS1><<<(ET + B - 1) / B, B, 0, stream>>>(ei, as1, ad1, m1);
  edge_sum_kernel<HEADS1><<<(ET + B - 1) / B, B, 0, stream>>>(ei, as1, ad1, m1, z1);
  {
    long long threads = (long long)ET * 32;             // one wave per edge
    int grid = (int)((threads + B - 1) / B);
    edge_agg1_kernel<<<grid, B, 0, stream>>>(ei, as1, ad1, m1, z1, h1, acc1);
  }
  relu_bias1_kernel<<<(N_NODES * HID_CH + B - 1) / B, B, 0, stream>>>(acc1, b1);

  // ---- layer 2 ----
  layer2_lin_kernel<<<(N_NODES + B - 1) / B, B, 0, stream>>>(acc1, W2, a_src2, a_dst2, h2, as2, ad2);
  edge_max_kernel<1><<<(ET + B - 1) / B, B, 0, stream>>>(ei, as2, ad2, m2);
  edge_sum_kernel<1><<<(ET + B - 1) / B, B, 0, stream>>>(ei, as2, ad2, m2, z2);
  edge_agg2_kernel<<<(ET + B - 1) / B, B, 0, stream>>>(ei, as2, ad2, m2, z2, h2, out);
  bias2_kernel<<<(N_NODES * OUT_CH + B - 1) / B, B, 0, stream>>>(out, b2);
}